// ComboLoss_73246372266596
// MI455X (gfx1250) — compile-verified
//
#include <hip/hip_runtime.h>
#include <hip/hip_bf16.h>
#include <math.h>

// ---------------------------------------------------------------------------
// ComboLoss (dice + BCE + boundary-EDT + focal) for pred/target (4,1,256,256).
// MI455X notes: problem is tiny (~2MB in, scalar out) -> latency bound.
// We fuse into 4 kernels, stage EDT rows in LDS via the gfx1250 async-to-LDS
// path, and do all sum-reductions on the matrix pipe via
// V_WMMA_F32_16X16X4_F32 with a ones B-matrix (wave32 sum = 2 WMMAs + 7 VALU
// adds). Deterministic two-stage reductions (no atomics).
// ---------------------------------------------------------------------------

#define BATCH 4
#define H 256
#define W 256
#define HW (H * W)          // 65536
#define NPIX (BATCH * HW)   // 262144
#define BIGD 1.0e6f

typedef float v2f __attribute__((ext_vector_type(2)));
typedef float v8f __attribute__((ext_vector_type(8)));

// ---- wave32 sum via WMMA f32 16x16x4 with ones matrix ----------------------
// A layout (32-bit A 16x4): vgpr0 = {lanes0-15: K=0, lanes16-31: K=2},
// vgpr1 = {K=1, K=3}. Set vgpr0 = v, vgpr1 = 0, B = ones:
//   D[m][n] = v_m + v_{m+16}.
// Lane-local add of the 8 D regs gives per-half partial; second WMMA folds the
// halves so every lane's D[0] is the full 32-lane sum.
__device__ __forceinline__ float wave_reduce_sum(float v) {
#if __has_builtin(__builtin_amdgcn_wmma_f32_16x16x4_f32)
  v2f a;    a.x = v;    a.y = 0.0f;
  v2f ones; ones.x = 1.0f; ones.y = 1.0f;
  v8f c = {};
  v8f d = __builtin_amdgcn_wmma_f32_16x16x4_f32(false, a, false, ones,
                                                (short)0, c, false, false);
  float w = ((d[0] + d[1]) + (d[2] + d[3])) + ((d[4] + d[5]) + (d[6] + d[7]));
  v2f a2; a2.x = w; a2.y = 0.0f;
  v8f d2 = __builtin_amdgcn_wmma_f32_16x16x4_f32(false, a2, false, ones,
                                                 (short)0, c, false, false);
  return d2[0];
#else
  for (int off = 16; off; off >>= 1) v += __shfl_xor(v, off, 32);
  return v;
#endif
}

// Block (256 threads = 8 waves) sum; valid on thread 0 only.
__device__ __forceinline__ float block_reduce_sum(float v, float* scratch8) {
  float t = wave_reduce_sum(v);
  const int lane = threadIdx.x & 31;
  const int wid  = threadIdx.x >> 5;
  __syncthreads();
  if (lane == 0) scratch8[wid] = t;
  __syncthreads();
  float s = 0.0f;
  if (threadIdx.x == 0) {
#pragma unroll
    for (int w = 0; w < 8; ++w) s += scratch8[w];
  }
  return s;
}

// ---------------------------------------------------------------------------
// Kernel 1: vertical 1D EDT (two scans) for both binarized images.
// blockIdx.x in [0,8): img = bi>>2 (0=pred,1=target), b = bi&3.
// Thread c owns column c -> every row access is a fully coalesced 1KB line.
// Writes g^2 (fp32) to workspace.
// ---------------------------------------------------------------------------
__global__ __launch_bounds__(256) void edt_pass1_kernel(
    const float* __restrict__ pred, const float* __restrict__ target,
    float* __restrict__ g2p, float* __restrict__ g2t) {
  const int img = blockIdx.x >> 2;
  const int b   = blockIdx.x & 3;
  const int c   = threadIdx.x;
  const float* src = (img == 0 ? pred : target) + b * HW;
  float*       g2  = (img == 0 ? g2p  : g2t)    + b * HW;

  // down scan: d[i] = (x==0) ? 0 : d[i-1]+1
  float carry = BIGD;
#pragma unroll 4
  for (int i = 0; i < H; ++i) {
    const float x = src[i * W + c];
    const bool zero = (img == 0) ? (x <= 0.5f) : (x == 0.0f);
    const float d = zero ? 0.0f : (carry + 1.0f);
    g2[i * W + c] = d;  // stash "down" temporarily
    carry = d;
  }
  // up scan + combine + square
  carry = BIGD;
#pragma unroll 4
  for (int i = H - 1; i >= 0; --i) {
    const float x = src[i * W + c];
    const bool zero = (img == 0) ? (x <= 0.5f) : (x == 0.0f);
    const float u = zero ? 0.0f : (carry + 1.0f);
    carry = u;
    const float g = fminf(g2[i * W + c], u);
    g2[i * W + c] = g * g;
  }
}

// ---------------------------------------------------------------------------
// Kernel 2: per-row min-plus pass  d2[j] = min_c g2[c] + (j-c)^2  for both
// images, then |sqrt(d2p)-sqrt(d2t)|, block-reduced to one partial per row.
// g2 rows staged in LDS via gfx1250 async-to-LDS (ASYNCcnt path).
// ---------------------------------------------------------------------------
__global__ __launch_bounds__(256) void edt_pass2_boundary_kernel(
    const float* __restrict__ g2p, const float* __restrict__ g2t,
    float* __restrict__ bd_part) {
  __shared__ float sp[W];
  __shared__ float st[W];
  __shared__ float red[8];

  const int row = blockIdx.x;        // b*H + i
  const int j   = threadIdx.x;
  const float* __restrict__ gp = g2p + row * W;
  const float* __restrict__ gt = g2t + row * W;

#if __has_builtin(__builtin_amdgcn_global_load_async_to_lds_b32)
  {
    typedef __attribute__((address_space(1))) int gas_int;
    typedef __attribute__((address_space(3))) int las_int;
    __builtin_amdgcn_global_load_async_to_lds_b32(
        (gas_int*)(gp + j), (las_int*)&sp[j], 0, 0);
    __builtin_amdgcn_global_load_async_to_lds_b32(
        (gas_int*)(gt + j), (las_int*)&st[j], 0, 0);
#if __has_builtin(__builtin_amdgcn_s_wait_asynccnt)
    __builtin_amdgcn_s_wait_asynccnt(0);
#else
    asm volatile("s_wait_asynccnt 0x0" ::: "memory");
#endif
  }
#else
  sp[j] = gp[j];
  st[j] = gt[j];
#endif
  __syncthreads();

  float mp = 3.0e38f, mt = 3.0e38f;
  float dd = (float)j;               // j - c, decremented each iteration
#pragma unroll 8
  for (int c = 0; c < W; ++c) {
    const float q = dd * dd;         // LDS reads below are wave broadcasts
    mp = fminf(mp, q + sp[c]);
    mt = fminf(mt, q + st[c]);
    dd -= 1.0f;
  }
  const float v = fabsf(sqrtf(mp) - sqrtf(mt));
  const float s = block_reduce_sum(v, red);
  if (threadIdx.x == 0) bd_part[row] = s;
}

// ---------------------------------------------------------------------------
// Kernel 3: elementwise dice/BCE/focal partial sums. 256 blocks x 256 thr,
// 4 elements each. Partials laid out as 5 arrays of 256 floats.
// ---------------------------------------------------------------------------
__global__ __launch_bounds__(256) void elementwise_kernel(
    const float* __restrict__ pred, const float* __restrict__ target,
    float* __restrict__ ew_part) {
  __shared__ float red[8];
  float s_p = 0.0f, s_t = 0.0f, s_pt = 0.0f, s_bce = 0.0f, s_foc = 0.0f;

  const int base = blockIdx.x * 1024 + threadIdx.x;
#pragma unroll
  for (int k = 0; k < 4; ++k) {
    const int idx = base + k * 256;
    const float x = pred[idx];
    const float t = target[idx];
    const float p = 1.0f / (1.0f + __expf(-x));
    // stable BCE-with-logits element == -(t*log p + (1-t)*log(1-p))
    const float bce = fmaxf(x, 0.0f) - x * t + log1pf(__expf(-fabsf(x)));
    const float pt  = p * t + (1.0f - p) * (1.0f - t);
    const float om  = 1.0f - pt;
    s_p   += p;
    s_t   += t;
    s_pt  += p * t;
    s_bce += bce;
    s_foc += om * om * bce;          // alpha=1, gamma=2
  }

  float r;
  r = block_reduce_sum(s_p,   red); if (threadIdx.x == 0) ew_part[0 * 256 + blockIdx.x] = r;
  r = block_reduce_sum(s_t,   red); if (threadIdx.x == 0) ew_part[1 * 256 + blockIdx.x] = r;
  r = block_reduce_sum(s_pt,  red); if (threadIdx.x == 0) ew_part[2 * 256 + blockIdx.x] = r;
  r = block_reduce_sum(s_bce, red); if (threadIdx.x == 0) ew_part[3 * 256 + blockIdx.x] = r;
  r = block_reduce_sum(s_foc, red); if (threadIdx.x == 0) ew_part[4 * 256 + blockIdx.x] = r;
}

// ---------------------------------------------------------------------------
// Kernel 4: deterministic final reduction (one block) + scalar loss.
// ---------------------------------------------------------------------------
__global__ __launch_bounds__(256) void finalize_kernel(
    const float* __restrict__ bd_part, const float* __restrict__ ew_part,
    float* __restrict__ out) {
  __shared__ float red[8];
  __shared__ float acc[6];
  const int t = threadIdx.x;

  // boundary: 1024 partials -> 4 per thread
  float v = bd_part[t] + bd_part[t + 256] + bd_part[t + 512] + bd_part[t + 768];
  float s = block_reduce_sum(v, red);
  if (t == 0) acc[5] = s;
#pragma unroll
  for (int i = 0; i < 5; ++i) {
    s = block_reduce_sum(ew_part[i * 256 + t], red);
    if (t == 0) acc[i] = s;
  }
  __syncthreads();

  if (t == 0) {
    const float N = (float)NPIX;
    const float s_p = acc[0], s_t = acc[1], s_pt = acc[2];
    const float s_bce = acc[3], s_foc = acc[4], s_bd = acc[5];
    const float dice = 1.0f - (2.0f * s_pt + 1.0f) / (s_p + s_t + 1.0f);
    const float bce  = s_bce / N;
    const float foc  = s_foc / N;
    const float bd   = s_bd / N;   // inner_boundary_weight = 1
    out[0] = 0.5f * dice + 0.5f * bce + 0.1f * bd + 0.2f * foc;
  }
}

// ---------------------------------------------------------------------------
extern "C" void kernel_launch(void* const* d_in, const int* in_sizes, int n_in,
                              void* d_out, int out_size, void* d_ws, size_t ws_size,
                              hipStream_t stream) {
  const float* pred   = (const float*)d_in[0];
  const float* target = (const float*)d_in[1];
  float* out = (float*)d_out;

  // workspace layout (floats):
  //   [0, 1024)                       bd_part
  //   [1024, 1024+5*256)              ew_part
  //   [4096, 4096+NPIX)               g2p
  //   [4096+NPIX, 4096+2*NPIX)        g2t
  float* ws      = (float*)d_ws;
  float* bd_part = ws;
  float* ew_part = ws + 1024;
  float* g2p     = ws + 4096;
  float* g2t     = ws + 4096 + NPIX;

  edt_pass1_kernel<<<8, 256, 0, stream>>>(pred, target, g2p, g2t);
  edt_pass2_boundary_kernel<<<BATCH * H, 256, 0, stream>>>(g2p, g2t, bd_part);
  elementwise_kernel<<<256, 256, 0, stream>>>(pred, target, ew_part);
  finalize_kernel<<<1, 256, 0, stream>>>(bd_part, ew_part, out);
}